// TGCNModel_21500606284424
// MI455X (gfx1250) — compile-verified
//
#include <hip/hip_runtime.h>

// Problem constants (from reference)
#define NN 50000
#define TT 12
#define EE 800000
// IN_C=32, HID=64, OUT_C=32

typedef __attribute__((ext_vector_type(2))) float v2f;
typedef __attribute__((ext_vector_type(8))) float v8f;

// ---------------------------------------------------------------------------
// Precompute folded gate matrices:
//   Mz[i][j] = sum_k Wz[i][k] * Lz_w[k][j]   (i<32, k<64, j<64; top half of Lz_w)
//   vz[j]    = sum_k bz[k]   * Lz_w[k][j] + Lz_b[j]
// (same for h). One block; tiny.
// ---------------------------------------------------------------------------
__global__ __launch_bounds__(256) void prep_kernel(
    const float* __restrict__ Wz, const float* __restrict__ bz,
    const float* __restrict__ Wh, const float* __restrict__ bh,
    const float* __restrict__ Lz_w, const float* __restrict__ Lz_b,
    const float* __restrict__ Lh_w, const float* __restrict__ Lh_b,
    float* __restrict__ Mz, float* __restrict__ vz,
    float* __restrict__ Mh, float* __restrict__ vh) {
  for (int idx = threadIdx.x; idx < 32 * 64; idx += blockDim.x) {
    const int i = idx >> 6, j = idx & 63;
    float sz = 0.f, sh = 0.f;
    for (int k = 0; k < 64; ++k) {
      sz += Wz[i * 64 + k] * Lz_w[k * 64 + j];
      sh += Wh[i * 64 + k] * Lh_w[k * 64 + j];
    }
    Mz[idx] = sz;
    Mh[idx] = sh;
  }
  for (int j = threadIdx.x; j < 64; j += blockDim.x) {
    float sz = Lz_b[j], sh = Lh_b[j];
    for (int k = 0; k < 64; ++k) {
      sz += bz[k] * Lz_w[k * 64 + j];
      sh += bh[k] * Lh_w[k * 64 + j];
    }
    vz[j] = sz;
    vh[j] = sh;
  }
}

// Zero the accumulator [N,64]
__global__ __launch_bounds__(256) void zero_acc_kernel(float* __restrict__ acc) {
  const size_t i = (size_t)blockIdx.x * blockDim.x + threadIdx.x;
  acc[i] = 0.0f;  // grid sized exactly N*64
}

// Per-timestep init: y[N*32]=0, deg[N]=1.0 (self-loop weight)
__global__ __launch_bounds__(256) void init_t_kernel(float* __restrict__ y,
                                                     float* __restrict__ deg) {
  const size_t i = (size_t)blockIdx.x * blockDim.x + threadIdx.x;
  y[i] = 0.0f;  // grid sized exactly N*32
  if (i < NN) deg[i] = 1.0f;
}

// deg[dst[e]] += w[e]
__global__ __launch_bounds__(256) void deg_kernel(const int* __restrict__ dst,
                                                  const float* __restrict__ ew,
                                                  float* __restrict__ deg) {
  const int e = blockIdx.x * blockDim.x + threadIdx.x;  // grid exactly E
  unsafeAtomicAdd(&deg[dst[e]], ew[e]);
}

// deg -> rsqrt(deg) in place
__global__ __launch_bounds__(256) void dinv_kernel(float* __restrict__ deg) {
  const int i = blockIdx.x * blockDim.x + threadIdx.x;
  if (i < NN) {
    const float d = deg[i];
    deg[i] = d > 0.0f ? rsqrtf(d) : 0.0f;
  }
}

// Edge scatter: one wave per edge, one lane per channel (32 channels).
// y[dst][c] += x[src][c] * dinv[src]*w*dinv[dst]
// x-row gather and y-row atomics are each a single contiguous 128B request.
__global__ __launch_bounds__(256) void scatter_kernel(
    const int* __restrict__ src, const int* __restrict__ dst,
    const float* __restrict__ ew, const float* __restrict__ dinv,
    const float* __restrict__ x, float* __restrict__ y) {
  const int e = blockIdx.x * 8 + (threadIdx.x >> 5);  // grid*8 == E exactly
  const int c = threadIdx.x & 31;
  const int s = src[e];
  const int d = dst[e];
  const float norm = dinv[s] * ew[e] * dinv[d];
  unsafeAtomicAdd(&y[(size_t)d * 32 + c], x[(size_t)s * 32 + c] * norm);
}

// ---------------------------------------------------------------------------
// Gate kernel (fp32 WMMA 16x16x4): per 16-node tile,
//   yfull = y + x * dinv^2   (self-loop folded in at A-load time)
//   Z  = sigmoid(yfull @ Mz + vz),  Ht = tanh(yfull @ Mh + vh)
//   acc += (1 - Z) * Ht
// A: 16x4 f32 (v2f/lane), B: 4x16 f32 (v2f/lane), C/D: 16x16 f32 (v8f/lane)
// ---------------------------------------------------------------------------
__global__ __launch_bounds__(256) void gate_wmma_kernel(
    const float* __restrict__ y, const float* __restrict__ x,
    const float* __restrict__ dinv,
    const float* __restrict__ Mz, const float* __restrict__ vz,
    const float* __restrict__ Mh, const float* __restrict__ vh,
    float* __restrict__ acc) {
  const int lane = threadIdx.x & 31;
  const int wave = threadIdx.x >> 5;
  const int tile = blockIdx.x * 8 + wave;
  if (tile * 16 >= NN) return;  // wave-uniform exit; EXEC stays all-1s
  const int node_base = tile * 16;
  const int row = lane & 15;
  const int khalf = (lane >> 4) << 1;  // 0 for lanes 0-15, 2 for lanes 16-31
  const int n = node_base + row;

  const float dv = dinv[n];
  const float dv2 = dv * dv;

  // A-tiles: 8 K-slabs of 16x4, with self-loop term fused.
  v2f a[8];
#pragma unroll
  for (int kt = 0; kt < 8; ++kt) {
    const int kk = kt * 4 + khalf;
    const float2 yv = *(const float2*)(y + (size_t)n * 32 + kk);
    const float2 xv = *(const float2*)(x + (size_t)n * 32 + kk);
    a[kt].x = yv.x + xv.x * dv2;
    a[kt].y = yv.y + xv.y * dv2;
  }

  const int rbase = node_base + ((lane >> 4) << 3);  // +8 rows for upper half
#pragma unroll
  for (int nt = 0; nt < 4; ++nt) {
    const int col = nt * 16 + (lane & 15);
    v8f cz = {};
    v8f ch = {};
#pragma unroll
    for (int kt = 0; kt < 8; ++kt) {
      const int kb = kt * 4 + khalf;
      v2f bzv, bhv;
      bzv.x = Mz[kb * 64 + col];
      bzv.y = Mz[(kb + 1) * 64 + col];
      bhv.x = Mh[kb * 64 + col];
      bhv.y = Mh[(kb + 1) * 64 + col];
      cz = __builtin_amdgcn_wmma_f32_16x16x4_f32(false, a[kt], false, bzv,
                                                 (short)0, cz, false, false);
      ch = __builtin_amdgcn_wmma_f32_16x16x4_f32(false, a[kt], false, bhv,
                                                 (short)0, ch, false, false);
    }
    const float bzc = vz[col];
    const float bhc = vh[col];
#pragma unroll
    for (int r = 0; r < 8; ++r) {
      const float zv = 1.0f / (1.0f + __expf(-(cz[r] + bzc)));
      const float hv = tanhf(ch[r] + bhc);
      float* p = acc + (size_t)(rbase + r) * 64 + col;
      *p += (1.0f - zv) * hv;  // exclusive owner per (node,col) this launch
    }
  }
}

// Final: out = (acc / T) @ W_out + b_out    [N,64]x[64,32]
__global__ __launch_bounds__(256) void out_wmma_kernel(
    const float* __restrict__ acc, const float* __restrict__ Wout,
    const float* __restrict__ bout, float* __restrict__ out) {
  const int lane = threadIdx.x & 31;
  const int wave = threadIdx.x >> 5;
  const int tile = blockIdx.x * 8 + wave;
  if (tile * 16 >= NN) return;
  const int node_base = tile * 16;
  const int row = lane & 15;
  const int khalf = (lane >> 4) << 1;
  const int n = node_base + row;
  const float scale = 1.0f / (float)TT;

  v2f a[16];
#pragma unroll
  for (int kt = 0; kt < 16; ++kt) {
    const int kk = kt * 4 + khalf;
    const float2 av = *(const float2*)(acc + (size_t)n * 64 + kk);
    a[kt].x = av.x * scale;
    a[kt].y = av.y * scale;
  }

  const int rbase = node_base + ((lane >> 4) << 3);
#pragma unroll
  for (int nt = 0; nt < 2; ++nt) {
    const int col = nt * 16 + (lane & 15);
    v8f c = {};
#pragma unroll
    for (int kt = 0; kt < 16; ++kt) {
      const int kb = kt * 4 + khalf;
      v2f bv;
      bv.x = Wout[kb * 32 + col];
      bv.y = Wout[(kb + 1) * 32 + col];
      c = __builtin_amdgcn_wmma_f32_16x16x4_f32(false, a[kt], false, bv,
                                                (short)0, c, false, false);
    }
    const float bc = bout[col];
#pragma unroll
    for (int r = 0; r < 8; ++r) {
      out[(size_t)(rbase + r) * 32 + col] = c[r] + bc;
    }
  }
}

// ---------------------------------------------------------------------------
extern "C" void kernel_launch(void* const* d_in, const int* in_sizes, int n_in,
                              void* d_out, int out_size, void* d_ws,
                              size_t ws_size, hipStream_t stream) {
  const float* x = (const float*)d_in[0];     // [T,N,32]
  const int* ei = (const int*)d_in[1];        // [T,2,E]
  const float* ew = (const float*)d_in[2];    // [T,E]
  const float* Wz = (const float*)d_in[3];
  const float* bz = (const float*)d_in[4];
  // d_in[5], d_in[6] (Wr, br): dead — R only multiplies H == 0
  const float* Wh = (const float*)d_in[7];
  const float* bh = (const float*)d_in[8];
  const float* Lz_w = (const float*)d_in[9];
  const float* Lz_b = (const float*)d_in[10];
  // d_in[11], d_in[12] (Lr_w, Lr_b): dead
  const float* Lh_w = (const float*)d_in[13];
  const float* Lh_b = (const float*)d_in[14];
  const float* Wout = (const float*)d_in[15];
  const float* bout = (const float*)d_in[16];
  float* out = (float*)d_out;

  // Workspace layout (floats), base assumed >= 256B aligned
  float* acc = (float*)d_ws;                 // N*64
  float* yb = acc + (size_t)NN * 64;         // N*32
  float* dinv = yb + (size_t)NN * 32;        // N   (deg, then rsqrt in place)
  float* Mz = dinv + NN;                     // 32*64
  float* Mh = Mz + 32 * 64;                  // 32*64
  float* vz = Mh + 32 * 64;                  // 64
  float* vh = vz + 64;                       // 64

  prep_kernel<<<1, 256, 0, stream>>>(Wz, bz, Wh, bh, Lz_w, Lz_b, Lh_w, Lh_b,
                                     Mz, vz, Mh, vh);
  zero_acc_kernel<<<(NN * 64) / 256, 256, 0, stream>>>(acc);

  for (int t = 0; t < TT; ++t) {
    const float* xt = x + (size_t)t * NN * 32;
    const int* src = ei + (size_t)t * 2 * EE;
    const int* dst = src + EE;
    const float* ewt = ew + (size_t)t * EE;

    init_t_kernel<<<(NN * 32) / 256, 256, 0, stream>>>(yb, dinv);
    deg_kernel<<<EE / 256, 256, 0, stream>>>(dst, ewt, dinv);
    dinv_kernel<<<(NN + 255) / 256, 256, 0, stream>>>(dinv);
    scatter_kernel<<<EE / 8, 256, 0, stream>>>(src, dst, ewt, dinv, xt, yb);
    gate_wmma_kernel<<<(NN / 16 + 7) / 8, 256, 0, stream>>>(yb, xt, dinv, Mz,
                                                            vz, Mh, vh, acc);
  }

  out_wmma_kernel<<<(NN / 16 + 7) / 8, 256, 0, stream>>>(acc, Wout, bout, out);
}